// PointNetFeaturePropagation_83588653515178
// MI455X (gfx1250) — compile-verified
//
#include <hip/hip_runtime.h>
#include <hip/hip_bf16.h>

// ---------------------------------------------------------------------------
// PointNet++ Feature Propagation for MI455X (gfx1250, wave32).
// fp32 end-to-end; conv layers on V_WMMA_F32_16X16X4_F32 with 16x64
// register blocking per wave (4 independent accumulators -> no WMMA RAW
// hazard stalls, 4x less A traffic).
// ---------------------------------------------------------------------------

typedef __attribute__((ext_vector_type(2))) float v2f;
typedef __attribute__((ext_vector_type(8))) float v8f;

#define B_    8
#define N_    8192
#define S_    2048
#define D1_   128
#define D2_   256
#define CIN0  384     // D1 + D2
#define C0_   256
#define C1_   128
#define KNN   16
#define ROWS  (B_ * N_)   // 65536

// ---------------------------------------------------------------------------
// Kernel 1: per-point 16-NN over S=2048 candidates (xyz2 staged in LDS),
// inverse-distance weights written to workspace.
// ---------------------------------------------------------------------------
__global__ __launch_bounds__(256)
void topk_kernel(const float* __restrict__ xyz1, const float* __restrict__ xyz2,
                 int* __restrict__ idxb, float* __restrict__ wb) {
  __shared__ float sx[S_ * 3];            // 24 KB of 320 KB LDS
  const int t = threadIdx.x;
  const int b = blockIdx.y;
  // vectorized stage: 6144 floats = 1536 float4s, 16B aligned per batch
  {
    const float4* src = (const float4*)(xyz2 + b * S_ * 3);
    float4* dst = (float4*)sx;
    for (int i = t; i < (S_ * 3) / 4; i += 256) dst[i] = src[i];
  }
  __syncthreads();

  const int n  = blockIdx.x * 256 + t;
  const long bn = (long)b * N_ + n;
  const float px = xyz1[bn * 3 + 0];
  const float py = xyz1[bn * 3 + 1];
  const float pz = xyz1[bn * 3 + 2];

  float bd[KNN]; int bi[KNN];
#pragma unroll
  for (int k = 0; k < KNN; ++k) { bd[k] = 3.4e38f; bi[k] = 0; }

  for (int s = 0; s < S_; ++s) {
    const float dx = px - sx[3 * s + 0];
    const float dy = py - sx[3 * s + 1];
    const float dz = pz - sx[3 * s + 2];
    const float d = dx * dx + dy * dy + dz * dz;
    if (d < bd[KNN - 1]) {                 // early-out keeps VALU cost low
      bd[KNN - 1] = d; bi[KNN - 1] = s;
#pragma unroll
      for (int j = KNN - 1; j > 0; --j) {  // compile-time indices -> registers
        if (bd[j] < bd[j - 1]) {
          float td = bd[j]; bd[j] = bd[j - 1]; bd[j - 1] = td;
          int   ti = bi[j]; bi[j] = bi[j - 1]; bi[j - 1] = ti;
        } else break;
      }
    }
  }

  float w[KNN]; float rs = 0.f;
#pragma unroll
  for (int k = 0; k < KNN; ++k) { w[k] = 1.f / (bd[k] + 1e-4f); rs += w[k]; }
  const float inv = 1.f / rs;
  const long base = bn * KNN;
#pragma unroll
  for (int k = 0; k < KNN; ++k) { idxb[base + k] = bi[k]; wb[base + k] = w[k] * inv; }
}

// ---------------------------------------------------------------------------
// Kernel 2: one block per query point; t<128 copies points1, t>=128 computes
// the weighted interpolation of points2 -> concat row of 384 channels.
// ---------------------------------------------------------------------------
__global__ __launch_bounds__(384)
void interp_concat_kernel(const float* __restrict__ points1,
                          const float* __restrict__ points2,
                          const int* __restrict__ idxb, const float* __restrict__ wb,
                          float* __restrict__ xcat) {
  __shared__ int   sidx[KNN];
  __shared__ float sw[KNN];
  const int t  = threadIdx.x;
  const int bn = blockIdx.x;
  const int b  = bn >> 13;                 // N = 8192
  if (t < KNN) { sidx[t] = idxb[(long)bn * KNN + t]; sw[t] = wb[(long)bn * KNN + t]; }
  __syncthreads();

  float* out = xcat + (long)bn * CIN0;
  if (t < D1_) {
    out[t] = points1[(long)bn * D1_ + t];
  } else {
    const int c = t - D1_;
    float acc = 0.f;
#pragma unroll
    for (int k = 0; k < KNN; ++k)
      acc += sw[k] * points2[((long)b * S_ + sidx[k]) * D2_ + c];
    out[t] = acc;
  }
}

// ---------------------------------------------------------------------------
// Kernel 3: fp32 GEMM on the tensor pipe, 16x64 register-blocked.
// Y[row, col] = sum_k X[row,k]*W[col,k] + bias[col]
// One wave = one 16x64 output tile: per K-step of 4 -> 1 A-frag load,
// 4 B-frag loads (same base, immediate offsets), 4 independent WMMAs.
// A frag (16x4 f32): lane<16 holds (M=lane, K=kk+0/1), lane>=16 (M, K=kk+2/3).
// B frag (4x16 f32): mirrored, N on lanes.  C/D: v8f per ISA layout.
// No divergence before WMMA -> EXEC all ones as required.
// ---------------------------------------------------------------------------
template <int CIN>
__global__ __launch_bounds__(256)
void gemm_bias_wmma_kernel(const float* __restrict__ X, const float* __restrict__ W,
                           const float* __restrict__ bias, float* __restrict__ Y,
                           const int NOUT) {
  const int lane  = threadIdx.x & 31;
  const int wave  = threadIdx.x >> 5;
  const int mtile = blockIdx.x * 8 + wave;  // 8 waves/block -> 128 rows/block
  const int row0  = mtile * 16;
  const int col0  = blockIdx.y * 64;        // 4 x 16 output columns per wave
  const int lo    = lane & 15;
  const int hi    = lane >> 4;

  v8f acc[4];
#pragma unroll
  for (int j = 0; j < 4; ++j) {
    const float bv = bias[col0 + 16 * j + lo];
    acc[j] = (v8f){bv, bv, bv, bv, bv, bv, bv, bv};
  }

  const float* arow = X + (size_t)(row0 + lo) * CIN + 2 * hi;
  const float* brow = W + (size_t)(col0 + lo) * CIN + 2 * hi;  // +j*16*CIN imm offs

  for (int kk = 0; kk < CIN; kk += 16) {
    // A stream is L2-resident (concat fits in 192MB L2); pull next K-slab
    // toward the WGP early.  Lowers to global_prefetch_b8.
    __builtin_prefetch(arow + kk + 16, 0, 1);
#pragma unroll
    for (int k4 = 0; k4 < 16; k4 += 4) {
      const v2f a = *(const v2f*)(arow + kk + k4);
#pragma unroll
      for (int j = 0; j < 4; ++j) {
        const v2f bfr = *(const v2f*)(brow + (size_t)j * 16 * CIN + kk + k4);
        acc[j] = __builtin_amdgcn_wmma_f32_16x16x4_f32(
            /*neg_a=*/false, a, /*neg_b=*/false, bfr,
            /*c_mod=*/(short)0, acc[j], /*reuse_a=*/false, /*reuse_b=*/false);
      }
    }
  }

  float* yout = Y + (size_t)(row0 + hi * 8) * NOUT + (col0 + lo);
#pragma unroll
  for (int j = 0; j < 4; ++j)
#pragma unroll
    for (int r = 0; r < 8; ++r) yout[(size_t)r * NOUT + 16 * j] = acc[j][r];
}

// ---------------------------------------------------------------------------
// Kernel 4: deterministic per-channel mean/var (training BN), fused into
// scale = gamma*rsqrt(var+eps), shift = beta - mean*scale.
// ---------------------------------------------------------------------------
__global__ __launch_bounds__(256)
void bn_stats_kernel(const float* __restrict__ Y, const int rows, const int C,
                     const float* __restrict__ gamma, const float* __restrict__ beta,
                     float* __restrict__ scale, float* __restrict__ shift) {
  __shared__ float ss[256], ssq[256];
  const int c = blockIdx.x;
  const int t = threadIdx.x;
  float s = 0.f, sq = 0.f;
  for (int r = t; r < rows; r += 256) {
    const float v = Y[(size_t)r * C + c];
    s += v; sq += v * v;
  }
  ss[t] = s; ssq[t] = sq;
  __syncthreads();
  for (int off = 128; off > 0; off >>= 1) {
    if (t < off) { ss[t] += ss[t + off]; ssq[t] += ssq[t + off]; }
    __syncthreads();
  }
  if (t == 0) {
    const float mean = ss[0] / (float)rows;
    const float var  = ssq[0] / (float)rows - mean * mean;
    const float rstd = rsqrtf(var + 1e-5f);
    const float sc   = gamma[c] * rstd;
    scale[c] = sc;
    shift[c] = beta[c] - mean * sc;
  }
}

// ---------------------------------------------------------------------------
// Kernel 5: y -> relu(y*scale + shift), float4-vectorized (C % 4 == 0, so the
// 4 lanes of a float4 stay within one row; c is 4-aligned for float4 params).
// ---------------------------------------------------------------------------
template <int C>
__global__ __launch_bounds__(256)
void bn_relu_kernel(const float* __restrict__ Y, float* __restrict__ out,
                    const long total4, const float* __restrict__ scale,
                    const float* __restrict__ shift) {
  const long i = (long)blockIdx.x * 256 + threadIdx.x;
  if (i < total4) {
    const int c4 = (int)((i * 4) & (C - 1));   // first channel of this float4
    const float4 y  = ((const float4*)Y)[i];
    const float4 sc = *(const float4*)(scale + c4);
    const float4 sh = *(const float4*)(shift + c4);
    float4 o;
    o.x = fmaxf(y.x * sc.x + sh.x, 0.f);
    o.y = fmaxf(y.y * sc.y + sh.y, 0.f);
    o.z = fmaxf(y.z * sc.z + sh.z, 0.f);
    o.w = fmaxf(y.w * sc.w + sh.w, 0.f);
    ((float4*)out)[i] = o;
  }
}

// ---------------------------------------------------------------------------
// Workspace layout (x1/y1 alias the dead concat region after GEMM1 consumed it):
//   [0, 4M)        neighbor indices
//   [4M, 8M)       neighbor weights
//   [8M, +100.66M) concat X0   (later: x1 [67.1M] | y1 [33.6M])
//   [OFF_Y0, +67M) y0
//   [OFF_ST, +3K)  BN scale/shift for both layers
// ---------------------------------------------------------------------------
static constexpr size_t OFF_IDX = 0;
static constexpr size_t OFF_WGT = OFF_IDX + (size_t)ROWS * KNN * sizeof(int);
static constexpr size_t OFF_CAT = OFF_WGT + (size_t)ROWS * KNN * sizeof(float);
static constexpr size_t OFF_Y0  = OFF_CAT + (size_t)ROWS * CIN0 * sizeof(float);
static constexpr size_t OFF_ST  = OFF_Y0 + (size_t)ROWS * C0_ * sizeof(float);

extern "C" void kernel_launch(void* const* d_in, const int* in_sizes, int n_in,
                              void* d_out, int out_size, void* d_ws, size_t ws_size,
                              hipStream_t stream) {
  const float* xyz1    = (const float*)d_in[0];
  const float* xyz2    = (const float*)d_in[1];
  const float* points1 = (const float*)d_in[2];
  const float* points2 = (const float*)d_in[3];
  const float* W0  = (const float*)d_in[4];
  const float* b0  = (const float*)d_in[5];
  const float* g0  = (const float*)d_in[6];
  const float* be0 = (const float*)d_in[7];
  const float* W1  = (const float*)d_in[8];
  const float* b1  = (const float*)d_in[9];
  const float* g1  = (const float*)d_in[10];
  const float* be1 = (const float*)d_in[11];

  char* ws = (char*)d_ws;
  int*   idxb  = (int*)(ws + OFF_IDX);
  float* wb    = (float*)(ws + OFF_WGT);
  float* xcat  = (float*)(ws + OFF_CAT);
  float* y0    = (float*)(ws + OFF_Y0);
  float* x1    = xcat;                                        // aliases concat
  float* y1    = (float*)(ws + OFF_CAT + (size_t)ROWS * C0_ * sizeof(float));
  float* scale0 = (float*)(ws + OFF_ST);
  float* shift0 = scale0 + C0_;
  float* scale1 = shift0 + C0_;
  float* shift1 = scale1 + C1_;

  // 1) 16-NN + inverse-distance weights
  topk_kernel<<<dim3(N_ / 256, B_), 256, 0, stream>>>(xyz1, xyz2, idxb, wb);

  // 2) interpolate + concat -> X0 [ROWS x 384]
  interp_concat_kernel<<<ROWS, 384, 0, stream>>>(points1, points2, idxb, wb, xcat);

  // 3) layer 0: GEMM (WMMA f32, 16x64/wave) -> BN stats -> BN+ReLU
  gemm_bias_wmma_kernel<CIN0><<<dim3(ROWS / 128, C0_ / 64), 256, 0, stream>>>(xcat, W0, b0, y0, C0_);
  bn_stats_kernel<<<C0_, 256, 0, stream>>>(y0, ROWS, C0_, g0, be0, scale0, shift0);
  bn_relu_kernel<C0_><<<(unsigned)(((long)ROWS * C0_ / 4 + 255) / 256), 256, 0, stream>>>(
      y0, x1, (long)ROWS * C0_ / 4, scale0, shift0);

  // 4) layer 1: GEMM (WMMA f32, 16x64/wave) -> BN stats -> BN+ReLU -> d_out
  gemm_bias_wmma_kernel<C0_><<<dim3(ROWS / 128, C1_ / 64), 256, 0, stream>>>(x1, W1, b1, y1, C1_);
  bn_stats_kernel<<<C1_, 256, 0, stream>>>(y1, ROWS, C1_, g1, be1, scale1, shift1);
  bn_relu_kernel<C1_><<<(unsigned)(((long)ROWS * C1_ / 4 + 255) / 256), 256, 0, stream>>>(
      y1, (float*)d_out, (long)ROWS * C1_ / 4, scale1, shift1);
}